// DistanceMatrix_29944511988234
// MI455X (gfx1250) — compile-verified
//
#include <hip/hip_runtime.h>
#include <hip/hip_bf16.h>
#include <math.h>

// Problem constants
#define NB 32      // batch
#define NL 512     // residues
#define NSTEP 1536 // 3*NL chain-extension steps
#define NF 32      // fragments per chain
#define SF 48      // steps per fragment (multiple of 3)
#define CAF 16     // C-alpha points per fragment (SF/3)

typedef float v2f __attribute__((ext_vector_type(2)));
typedef float v8f __attribute__((ext_vector_type(8)));

// ---------------------------------------------------------------------------
// Kernel 1a: simulate each 48-step fragment independently from canonical init.
// One thread per (batch, fragment): 32*32 = 1024 threads.
// ---------------------------------------------------------------------------
__global__ __launch_bounds__(256) void frag_sim(const float* __restrict__ tors,
                                                float* __restrict__ fragPts,
                                                float* __restrict__ triples) {
  int tid = blockIdx.x * blockDim.x + threadIdx.x;
  if (tid >= NB * NF) return;
  int b = tid >> 5;
  int f = tid & (NF - 1);

  // bond geometry: r*cos(pi-theta), r*sin(pi-theta)
  const float BL[3] = {145.801f, 152.326f, 132.868f};
  const float BA[3] = {2.124f, 1.941f, 2.028f};
  float rc[3], rs[3];
#pragma unroll
  for (int k = 0; k < 3; ++k) {
    float th = 3.14159265358979f - BA[k];
    float s, c;
    __sincosf(th, &s, &c);
    rc[k] = BL[k] * c;
    rs[k] = BL[k] * s;
  }

  // canonical init triple (same as reference)
  float ax = -0.70710678f, ay = 1.22474487f, az = 0.f;
  float bx = -1.41421356f, by = 0.f, bz = 0.f;
  float cx = 0.f, cy = 0.f, cz = 0.f;

  const float* tb = tors + (size_t)b * NSTEP;  // tors[b][l][k] flat == tors[b*1536 + t]
  float* outp = fragPts + (size_t)tid * CAF * 3;

  int k = 0;  // fragment start is multiple of 3 -> k cycles 0,1,2
  for (int t = f * SF; t < f * SF + SF; ++t) {
    float d = tb[t];
    float sd, cd;
    __sincosf(d, &sd, &cd);
    float px = rc[k], py = cd * rs[k], pz = sd * rs[k];

    float bcx = cx - bx, bcy = cy - by, bcz = cz - bz;
    float inv = __builtin_amdgcn_rsqf(bcx * bcx + bcy * bcy + bcz * bcz);
    bcx *= inv; bcy *= inv; bcz *= inv;
    float bax = bx - ax, bay = by - ay, baz = bz - az;
    float nx = bay * bcz - baz * bcy;
    float ny = baz * bcx - bax * bcz;
    float nz = bax * bcy - bay * bcx;
    inv = __builtin_amdgcn_rsqf(nx * nx + ny * ny + nz * nz);
    nx *= inv; ny *= inv; nz *= inv;
    float mx = ny * bcz - nz * bcy;
    float my = nz * bcx - nx * bcz;
    float mz = nx * bcy - ny * bcx;
    float wx = px * bcx + py * mx + pz * nx + cx;
    float wy = px * bcy + py * my + pz * ny + cy;
    float wz = px * bcz + py * mz + pz * nz + cz;
    ax = bx; ay = by; az = bz;
    bx = cx; by = cy; bz = cz;
    cx = wx; cy = wy; cz = wz;
    if (k == 1) {  // C-alpha
      outp[0] = wx; outp[1] = wy; outp[2] = wz;
      outp += 3;
    }
    k = (k == 2) ? 0 : k + 1;
  }
  float* tr = triples + (size_t)tid * 9;
  tr[0] = ax; tr[1] = ay; tr[2] = az;
  tr[3] = bx; tr[4] = by; tr[5] = bz;
  tr[6] = cx; tr[7] = cy; tr[8] = cz;
}

// frame from triple: columns [bc, m1, n], row-major 3x3
__device__ inline void make_frame(float ax, float ay, float az,
                                  float bx, float by, float bz,
                                  float cx, float cy, float cz, float F[9]) {
  float bcx = cx - bx, bcy = cy - by, bcz = cz - bz;
  float inv = __builtin_amdgcn_rsqf(bcx * bcx + bcy * bcy + bcz * bcz);
  bcx *= inv; bcy *= inv; bcz *= inv;
  float bax = bx - ax, bay = by - ay, baz = bz - az;
  float nx = bay * bcz - baz * bcy;
  float ny = baz * bcx - bax * bcz;
  float nz = bax * bcy - bay * bcx;
  inv = __builtin_amdgcn_rsqf(nx * nx + ny * ny + nz * nz);
  nx *= inv; ny *= inv; nz *= inv;
  float mx = ny * bcz - nz * bcy;
  float my = nz * bcx - nx * bcz;
  float mz = nx * bcy - ny * bcx;
  F[0] = bcx; F[1] = mx; F[2] = nx;
  F[3] = bcy; F[4] = my; F[5] = ny;
  F[6] = bcz; F[7] = mz; F[8] = nz;
}

// ---------------------------------------------------------------------------
// Kernel 1b: per batch, sequentially compose the 32 fragment transforms.
// ---------------------------------------------------------------------------
__global__ void stitch(const float* __restrict__ triples, float* __restrict__ X) {
  int b = blockIdx.x * blockDim.x + threadIdx.x;
  if (b >= NB) return;

  float F0[9];
  make_frame(-0.70710678f, 1.22474487f, 0.f, -1.41421356f, 0.f, 0.f, 0.f, 0.f, 0.f, F0);

  float R[9] = {1.f, 0.f, 0.f, 0.f, 1.f, 0.f, 0.f, 0.f, 1.f};
  float tx = 0.f, ty = 0.f, tz = 0.f;

  for (int f = 0; f < NF; ++f) {
    float* dst = X + ((size_t)b * NF + f) * 12;
#pragma unroll
    for (int i = 0; i < 9; ++i) dst[i] = R[i];
    dst[9] = tx; dst[10] = ty; dst[11] = tz;

    const float* tr = triples + ((size_t)b * NF + f) * 9;
    float FL[9];
    make_frame(tr[0], tr[1], tr[2], tr[3], tr[4], tr[5], tr[6], tr[7], tr[8], FL);
    // Rf = FL * F0^T
    float Rf[9];
#pragma unroll
    for (int i = 0; i < 3; ++i)
#pragma unroll
      for (int j = 0; j < 3; ++j)
        Rf[i * 3 + j] = FL[i * 3 + 0] * F0[j * 3 + 0] + FL[i * 3 + 1] * F0[j * 3 + 1] +
                        FL[i * 3 + 2] * F0[j * 3 + 2];
    // t_new = R * c_last + t ; R_new = R * Rf
    float ntx = R[0] * tr[6] + R[1] * tr[7] + R[2] * tr[8] + tx;
    float nty = R[3] * tr[6] + R[4] * tr[7] + R[5] * tr[8] + ty;
    float ntz = R[6] * tr[6] + R[7] * tr[7] + R[8] * tr[8] + tz;
    float Rn[9];
#pragma unroll
    for (int i = 0; i < 3; ++i)
#pragma unroll
      for (int j = 0; j < 3; ++j)
        Rn[i * 3 + j] = R[i * 3 + 0] * Rf[0 * 3 + j] + R[i * 3 + 1] * Rf[1 * 3 + j] +
                        R[i * 3 + 2] * Rf[2 * 3 + j];
#pragma unroll
    for (int i = 0; i < 9; ++i) R[i] = Rn[i];
    tx = ntx; ty = nty; tz = ntz;
  }
}

// ---------------------------------------------------------------------------
// Kernel 1c: apply transforms, emit coords output + padded X[.][4] + norms.
// ---------------------------------------------------------------------------
__global__ __launch_bounds__(256) void apply_xform(const float* __restrict__ fragPts,
                                                   const float* __restrict__ T,
                                                   float* __restrict__ coords,
                                                   float* __restrict__ Xpad,
                                                   float* __restrict__ norms) {
  int idx = blockIdx.x * blockDim.x + threadIdx.x;
  if (idx >= NB * NL) return;
  int b = idx >> 9;
  int l = idx & (NL - 1);
  int f = l >> 4;       // 16 C-alphas per fragment
  int i = l & (CAF - 1);
  const float* Tp = T + ((size_t)b * NF + f) * 12;
  const float* p = fragPts + (((size_t)b * NF + f) * CAF + i) * 3;
  float x = Tp[0] * p[0] + Tp[1] * p[1] + Tp[2] * p[2] + Tp[9];
  float y = Tp[3] * p[0] + Tp[4] * p[1] + Tp[5] * p[2] + Tp[10];
  float z = Tp[6] * p[0] + Tp[7] * p[1] + Tp[8] * p[2] + Tp[11];
  coords[(size_t)idx * 3 + 0] = x;
  coords[(size_t)idx * 3 + 1] = y;
  coords[(size_t)idx * 3 + 2] = z;
  float4 v = make_float4(x, y, z, 0.f);
  ((float4*)Xpad)[idx] = v;
  norms[idx] = x * x + y * y + z * z;
}

// ---------------------------------------------------------------------------
// Kernel 2: distance matrix via V_WMMA_F32_16X16X4_F32.
// One wave per (batch, 16-row stripe): 1024 waves. A fragment loaded once per
// wave; the 32 column tiles are processed unrolled x4 so 4 WMMAs + their B
// loads are in flight. Epilogue: d = v_sqrt(v_max(ni + nj - 2G, 0)).
// A layout (ISA 16x4): v0 = K0|K2 (lane<16|>=16), v1 = K1|K3, row = lane&15.
// C/D layout: VGPR r -> M = r + 8*(lane>=16), N = lane&15.
// ---------------------------------------------------------------------------
__global__ __launch_bounds__(256) void dist_wmma(const float* __restrict__ Xpad,
                                                 const float* __restrict__ norms,
                                                 float* __restrict__ dist) {
  const int lane = threadIdx.x & 31;
  const int wid = (blockIdx.x * blockDim.x + threadIdx.x) >> 5;  // 0..1023
  const int b = wid >> 5;
  const int i0 = (wid & 31) << 4;
  const int hi = lane >> 4;  // 0: K0/K1, M+0 ; 1: K2/K3, M+8
  const int lo = lane & 15;

  const float* Xb = Xpad + (size_t)b * NL * 4;
  const float* Nb = norms + (size_t)b * NL;
  float* Db = dist + (size_t)b * NL * NL;

  // A fragment: rows i0+lo, two K values per lane
  v2f a = ((const v2f*)(Xb + (size_t)(i0 + lo) * 4))[hi];

  float rn[8];
#pragma unroll
  for (int r = 0; r < 8; ++r) rn[r] = Nb[i0 + hi * 8 + r];

#pragma unroll 4
  for (int j0 = 0; j0 < NL; j0 += 16) {
    v2f bf = ((const v2f*)(Xb + (size_t)(j0 + lo) * 4))[hi];
    float cn = Nb[j0 + lo];
    v8f acc = {};
    acc = __builtin_amdgcn_wmma_f32_16x16x4_f32(
        /*neg_a=*/false, a, /*neg_b=*/false, bf,
        /*c_mod=*/(short)0, acc, /*reuse_a=*/false, /*reuse_b=*/false);
#pragma unroll
    for (int r = 0; r < 8; ++r) {
      float s = rn[r] + cn - 2.0f * acc[r];
      s = fmaxf(s, 0.0f);                       // where(sq>0, ., 0) semantics
      float d = __builtin_amdgcn_sqrtf(s);      // raw v_sqrt_f32
      Db[(size_t)(i0 + hi * 8 + r) * NL + (j0 + lo)] = d;
    }
  }
}

// ---------------------------------------------------------------------------
extern "C" void kernel_launch(void* const* d_in, const int* in_sizes, int n_in,
                              void* d_out, int out_size, void* d_ws, size_t ws_size,
                              hipStream_t stream) {
  (void)in_sizes; (void)n_in; (void)out_size; (void)ws_size;
  const float* tors = (const float*)d_in[0];  // [32,512,3] f32
  float* out = (float*)d_out;
  float* dist = out;                               // [32,512,512]
  float* coords = out + (size_t)NB * NL * NL;      // [32,512,3]

  float* ws = (float*)d_ws;
  float* fragPts = ws;                                   // 32*32*16*3 = 49152
  float* triples = ws + 49152;                           // 32*32*9   =  9216
  float* xforms  = ws + 49152 + 9216;                    // 32*32*12  = 12288
  float* Xpad    = ws + 49152 + 9216 + 12288;            // 32*512*4  = 65536
  float* nrm     = ws + 49152 + 9216 + 12288 + 65536;    // 32*512    = 16384

  frag_sim<<<(NB * NF + 255) / 256, 256, 0, stream>>>(tors, fragPts, triples);
  stitch<<<1, 32, 0, stream>>>(triples, xforms);
  apply_xform<<<(NB * NL + 255) / 256, 256, 0, stream>>>(fragPts, xforms, coords, Xpad, nrm);
  dist_wmma<<<(NB * NF * 32 + 255) / 256, 256, 0, stream>>>(Xpad, nrm, dist);
}